// LandmarkDecoder_41721312314237
// MI455X (gfx1250) — compile-verified
//
#include <hip/hip_runtime.h>
#include <hip/hip_bf16.h>

typedef __attribute__((ext_vector_type(2))) float v2f;
typedef __attribute__((ext_vector_type(8))) float v8f;
typedef int v4i __attribute__((vector_size(16)));   // matches builtin proto

#define D128 128
#define HEADS 4
#define SEQ 2048
#define BATCH 512
#define CH 64           // s-positions per streamed chunk
#define PADR 132        // padded LDS row stride (floats)

#if __has_builtin(__builtin_amdgcn_global_load_async_to_lds_b128) && \
    __has_builtin(__builtin_amdgcn_s_wait_asynccnt)
#define USE_ASYNC_LDS 1
#else
#define USE_ASYNC_LDS 0
#endif

// 16-byte global->LDS copy: async (direct-to-LDS, ASYNCcnt) when available.
// Builtin signature (probe-confirmed via diagnostic):
//   (v4i addrspace(1)* src_global, v4i addrspace(3)* dst_lds, imm off, imm cpol)
__device__ __forceinline__ void cp16_g2l(const float* __restrict__ g,
                                         float* __restrict__ l) {
#if USE_ASYNC_LDS
  __builtin_amdgcn_global_load_async_to_lds_b128(
      (__attribute__((address_space(1))) v4i*)(v4i*)const_cast<float*>(g),
      (__attribute__((address_space(3))) v4i*)(v4i*)l, 0, 0);
#else
  *(float4*)l = *(const float4*)g;
#endif
}

__device__ __forceinline__ void wait_async_done() {
#if USE_ASYNC_LDS
  __builtin_amdgcn_s_wait_asynccnt(0);
#endif
}

// ---------------------------------------------------------------------------
// WMMA f32 GEMM: Y[M,N] = X[M,K] @ W[N,K]^T + bias[N], optional ReLU.
// One wave per 16x16 tile, K stepped by 4 via V_WMMA_F32_16X16X4_F32.
// Requires M%16==0, N%16==0, K%4==0.
// ---------------------------------------------------------------------------
template <int RELU>
__global__ __launch_bounds__(32)
void gemm_wmma_f32(const float* __restrict__ X, const float* __restrict__ W,
                   const float* __restrict__ bias, float* __restrict__ Y,
                   int M, int N, int K) {
  const int ntn = N >> 4;
  const int tn = (blockIdx.x % ntn) << 4;
  const int tm = (blockIdx.x / ntn) << 4;
  const int lane = threadIdx.x & 31;
  const int half = lane >> 4;      // 0: K pair {k,k+1}, 1: {k+2,k+3}
  const int l16  = lane & 15;
  v8f acc = {0.f, 0.f, 0.f, 0.f, 0.f, 0.f, 0.f, 0.f};
  const float* __restrict__ xr = X + (size_t)(tm + l16) * K + 2 * half;
  const float* __restrict__ wr = W + (size_t)(tn + l16) * K + 2 * half;
  for (int k = 0; k < K; k += 4) {
    v2f a; a.x = xr[k]; a.y = xr[k + 1];
    v2f b; b.x = wr[k]; b.y = wr[k + 1];
    acc = __builtin_amdgcn_wmma_f32_16x16x4_f32(false, a, false, b,
                                                (short)0, acc, false, false);
  }
  const int n = tn + l16;
  const float bn = bias[n];
#pragma unroll
  for (int r = 0; r < 8; ++r) {
    float v = acc[r] + bn;
    if (RELU) v = fmaxf(v, 0.f);
    Y[(size_t)(tm + r + 8 * half) * N + n] = v;
  }
}

// ---------------------------------------------------------------------------
// Parameter folding: P[(h,d),e] = scale * sum_j Wk[h*32+j,d]*Wq[h*32+j,e]
//                    r[(h,d)]   = scale * sum_j Wk[h*32+j,d]*bq[h*32+j]
// (bk drops out: s-independent logit offset is softmax-invariant)
// ---------------------------------------------------------------------------
__global__ __launch_bounds__(256)
void prep_avec_mat(const float* __restrict__ Wk, const float* __restrict__ Wq,
                   const float* __restrict__ bq, float* __restrict__ P,
                   float* __restrict__ r) {
  const int t = blockIdx.x * 256 + threadIdx.x;   // 65536 threads
  const int e = t & 127;
  const int hd = t >> 7;                          // 0..511
  const int h = hd >> 7, d = hd & 127;
  const float scale = 0.17677669529663687f;       // 1/sqrt(32)
  float s = 0.f;
#pragma unroll 4
  for (int j = 0; j < 32; ++j) {
    const int row = h * 32 + j;
    s += Wk[row * D128 + d] * Wq[row * D128 + e];
  }
  P[hd * D128 + e] = s * scale;
  if (e == 0) {
    float rr = 0.f;
    for (int j = 0; j < 32; ++j) {
      const int row = h * 32 + j;
      rr += Wk[row * D128 + d] * bq[row];
    }
    r[hd] = rr * scale;
  }
}

// Mc[o,(h,d)] = sum_j Wo[o,h*32+j]*Wv[h*32+j,d];  bc[o] = Wo[o,:]·bv + bo[o]
__global__ __launch_bounds__(256)
void prep_out_mat(const float* __restrict__ Wo, const float* __restrict__ Wv,
                  const float* __restrict__ bv, const float* __restrict__ bo,
                  float* __restrict__ Mc, float* __restrict__ bc) {
  const int t = blockIdx.x * 256 + threadIdx.x;   // 65536 threads
  const int hd = t & 511;
  const int o = t >> 9;                           // 0..127
  const int h = hd >> 7, d = hd & 127;
  float s = 0.f;
#pragma unroll 4
  for (int j = 0; j < 32; ++j)
    s += Wo[o * D128 + h * 32 + j] * Wv[(h * 32 + j) * D128 + d];
  Mc[o * 512 + hd] = s;
  if (hd == 0) {
    float s2 = bo[o];
    for (int k = 0; k < D128; ++k) s2 += Wo[o * D128 + k] * bv[k];
    bc[o] = s2;
  }
}

// ---------------------------------------------------------------------------
// HBM-bound streaming attention (single-query flash softmax), double-buffered
// through LDS with async direct-to-LDS loads (ASYNCcnt) when available.
// One block per batch; streams X[b,s,:] exactly once.
// ctx[b,h,:] = sum_s softmax_s(a[b,h,:]·x_s) * x_s
// ---------------------------------------------------------------------------
__global__ __launch_bounds__(256)
void attn_stream(const float* __restrict__ X,    // (B,S,128)
                 const float* __restrict__ AV,   // (B,4,128) logit vectors
                 float* __restrict__ CTX,        // (B,4,128) normalized ctx
                 int S) {
  __shared__ __align__(16) float xs[2][CH * PADR];
  __shared__ __align__(16) float av[HEADS * D128];
  __shared__ float wb[CH * HEADS];   // logits -> exp weights, [s*4+h]
  __shared__ float ps[8];            // per-wave reduction scratch

  const int b = blockIdx.x;
  const int t = threadIdx.x;
  const int lane = t & 31, wave = t >> 5;
  const int sl = t & 63;             // s-local (logit phase)
  const int hh = t >> 6;             // head   (logit phase)
  const int d  = t & 127;            // dim    (ctx phase)
  const int h0 = (t >> 7) * 2;       // head pair (ctx phase)

  for (int i = t; i < HEADS * D128; i += 256) av[i] = AV[(size_t)b * 512 + i];

  float m[HEADS] = {-1e30f, -1e30f, -1e30f, -1e30f};
  float l[HEADS] = {0.f, 0.f, 0.f, 0.f};
  float acc0 = 0.f, acc1 = 0.f;
  const float4* __restrict__ av4 = (const float4*)(av + hh * D128);
  const float* __restrict__ Xb = X + (size_t)b * S * D128;
  const int nch = S / CH;

  // prologue: stage chunk 0 into buffer 0 (each wave: one row per step)
#pragma unroll
  for (int j = 0; j < (CH * 32) / 256; ++j) {
    const int i = t + j * 256;
    cp16_g2l(Xb + 4 * i, &xs[0][(i >> 5) * PADR + 4 * (i & 31)]);
  }
  wait_async_done();
  __syncthreads();   // chunk 0 + av visible to all waves

  int buf = 0;
  for (int c = 0; c < nch; ++c) {
    // --- issue async loads for next chunk into the other buffer ---
    if (c + 1 < nch) {
      const float* __restrict__ Xg = Xb + (size_t)(c + 1) * CH * D128;
#pragma unroll
      for (int j = 0; j < (CH * 32) / 256; ++j) {
        const int i = t + j * 256;
        cp16_g2l(Xg + 4 * i, &xs[buf ^ 1][(i >> 5) * PADR + 4 * (i & 31)]);
      }
#if !USE_ASYNC_LDS
      // fallback path: help the sync loads along
      __builtin_prefetch(Xg + t * 32, 0, 0);
#endif
    }
    const float* __restrict__ xcur = xs[buf];

    // --- logits: thread (sl,hh) computes full 128-dim dot ---
    {
      const float4* __restrict__ xr = (const float4*)(xcur + sl * PADR);
      float a = 0.f;
#pragma unroll
      for (int q = 0; q < 32; ++q) {
        const float4 xv = xr[q], aw = av4[q];
        a += xv.x * aw.x + xv.y * aw.y + xv.z * aw.z + xv.w * aw.w;
      }
      wb[sl * 4 + hh] = a;
      float wm = a;
#pragma unroll
      for (int off = 16; off; off >>= 1) wm = fmaxf(wm, __shfl_xor(wm, off));
      if (lane == 0) ps[wave] = wm;   // waves 2h,2h+1 cover head h
    }
    __syncthreads();

    float mn[HEADS], f[HEADS];
#pragma unroll
    for (int h = 0; h < HEADS; ++h) {
      mn[h] = fmaxf(m[h], fmaxf(ps[2 * h], ps[2 * h + 1]));
      f[h] = __expf(m[h] - mn[h]);    // accumulator rescale factor
    }
    __syncthreads();   // everyone consumed ps before it is overwritten

    // --- exp weights + per-head exp-sum ---
    {
      const float e = __expf(wb[sl * 4 + hh] - mn[hh]);
      wb[sl * 4 + hh] = e;
      float es = e;
#pragma unroll
      for (int off = 16; off; off >>= 1) es += __shfl_xor(es, off);
      if (lane == 0) ps[wave] = es;
    }
    __syncthreads();

#pragma unroll
    for (int h = 0; h < HEADS; ++h) {
      l[h] = l[h] * f[h] + (ps[2 * h] + ps[2 * h + 1]);
      m[h] = mn[h];
    }
    acc0 *= f[h0];
    acc1 *= f[h0 + 1];

    // --- ctx accumulation: thread owns (h0,d) and (h0+1,d) ---
#pragma unroll 8
    for (int s = 0; s < CH; ++s) {
      const float xv = xcur[s * PADR + d];
      acc0 += wb[s * 4 + h0] * xv;
      acc1 += wb[s * 4 + h0 + 1] * xv;
    }

    // next buffer fully resident + wb/xs reuse safe before next iteration
    wait_async_done();
    __syncthreads();
    buf ^= 1;
  }

  CTX[(size_t)b * 512 + h0 * D128 + d]       = acc0 / l[h0];
  CTX[(size_t)b * 512 + (h0 + 1) * D128 + d] = acc1 / l[h0 + 1];
}

// pad f2b weights 182x128 -> 192x128 (zeros) so the last GEMM stays on WMMA
__global__ __launch_bounds__(256)
void pad_w192(const float* __restrict__ W, const float* __restrict__ bsrc,
              float* __restrict__ Wp, float* __restrict__ bp) {
  const int t = blockIdx.x * 256 + threadIdx.x;   // 24576
  const int n = t >> 7, k = t & 127;
  Wp[t] = (n < 182) ? W[n * 128 + k] : 0.f;
  if (k == 0) bp[n] = (n < 182) ? bsrc[n] : 0.f;
}

// out[b,0,p,c]: p<40 from gt_lm, p in [40,130] from nml (stride-192 rows)
__global__ __launch_bounds__(256)
void scatter_out(const float* __restrict__ gt, const float* __restrict__ nml,
                 float* __restrict__ out) {
  const int t = blockIdx.x * 256 + threadIdx.x;   // 512*262 = 134144 exact
  const int b = t / 262, i = t % 262;
  const int p = i >> 1, c = i & 1;
  out[t] = (p < 40) ? gt[b * 262 + i] : nml[b * 192 + (p - 40) * 2 + c];
}

// ---------------------------------------------------------------------------
extern "C" void kernel_launch(void* const* d_in, const int* in_sizes, int n_in,
                              void* d_out, int out_size, void* d_ws, size_t ws_size,
                              hipStream_t stream) {
  (void)in_sizes; (void)n_in; (void)out_size; (void)ws_size;
  // Input order: insertion order for top dict, sorted-key pytree for params.
  const float* audio = (const float*)d_in[0];
  const float* llfs  = (const float*)d_in[1];
  const float* lmf   = (const float*)d_in[2];
  const float* gt    = (const float*)d_in[3];
  const float* Wf1a = (const float*)d_in[4],  *bf1a = (const float*)d_in[5];
  const float* Wf1b = (const float*)d_in[6],  *bf1b = (const float*)d_in[7];
  const float* Wf2a = (const float*)d_in[8],  *bf2a = (const float*)d_in[9];
  const float* Wf2b = (const float*)d_in[10], *bf2b = (const float*)d_in[11];
  // mha dicts sorted: Wk, Wo, Wq, Wv, bk, bo, bq, bv
  const float* Wk1 = (const float*)d_in[12], *Wo1 = (const float*)d_in[13];
  const float* Wq1 = (const float*)d_in[14], *Wv1 = (const float*)d_in[15];
  /* bk1 unused (softmax-invariant) */        const float* bo1 = (const float*)d_in[17];
  const float* bq1 = (const float*)d_in[18], *bv1 = (const float*)d_in[19];
  const float* Wk3 = (const float*)d_in[20], *Wo3 = (const float*)d_in[21];
  const float* Wq3 = (const float*)d_in[22], *Wv3 = (const float*)d_in[23];
  const float* bo3 = (const float*)d_in[25];
  const float* bq3 = (const float*)d_in[26], *bv3 = (const float*)d_in[27];
  float* out = (float*)d_out;

  float* ws = (float*)d_ws;
  float* P1  = ws;            ws += 512 * 128;
  float* r1  = ws;            ws += 512;
  float* M1  = ws;            ws += 128 * 512;
  float* bc1 = ws;            ws += 128;
  float* P3  = ws;            ws += 512 * 128;
  float* r3  = ws;            ws += 512;
  float* M3  = ws;            ws += 128 * 512;
  float* bc3 = ws;            ws += 128;
  float* AV  = ws;            ws += 512 * 512;
  float* CT  = ws;            ws += 512 * 512;
  float* B0  = ws;            ws += 512 * 128;
  float* B1  = ws;            ws += 512 * 128;
  float* B2  = ws;            ws += 512 * 192;
  float* W2p = ws;            ws += 192 * 128;
  float* b2p = ws;            ws += 256;

  // parameter folding (tiny, parameter-only)
  prep_avec_mat<<<256, 256, 0, stream>>>(Wk1, Wq1, bq1, P1, r1);
  prep_avec_mat<<<256, 256, 0, stream>>>(Wk3, Wq3, bq3, P3, r3);
  prep_out_mat<<<256, 256, 0, stream>>>(Wo1, Wv1, bv1, bo1, M1, bc1);
  prep_out_mat<<<256, 256, 0, stream>>>(Wo3, Wv3, bv3, bo3, M3, bc3);
  pad_w192<<<96, 256, 0, stream>>>(Wf2b, bf2b, W2p, b2p);

  // avec1 = lm @ P1^T + r1   (512x512x128, WMMA)
  gemm_wmma_f32<0><<<32 * 32, 32, 0, stream>>>(lmf, P1, r1, AV, 512, 512, 128);
  // HBM-bound pass #1: audio (512 MB, read once)
  attn_stream<<<BATCH, 256, 0, stream>>>(audio, AV, CT, SEQ);
  // lm = ctx @ M1^T + bc1    (512x128x512, WMMA)
  gemm_wmma_f32<0><<<32 * 8, 32, 0, stream>>>(CT, M1, bc1, B0, 512, 128, 512);
  // MLP f1
  gemm_wmma_f32<1><<<32 * 8, 32, 0, stream>>>(B0, Wf1a, bf1a, B1, 512, 128, 128);
  gemm_wmma_f32<0><<<32 * 8, 32, 0, stream>>>(B1, Wf1b, bf1b, B0, 512, 128, 128);
  // avec3 = f1 @ P3^T + r3
  gemm_wmma_f32<0><<<32 * 32, 32, 0, stream>>>(B0, P3, r3, AV, 512, 512, 128);
  // HBM-bound pass #2: llfs (512 MB, read once)
  attn_stream<<<BATCH, 256, 0, stream>>>(llfs, AV, CT, SEQ);
  // f2 = ctx @ M3^T + bc3
  gemm_wmma_f32<0><<<32 * 8, 32, 0, stream>>>(CT, M3, bc3, B1, 512, 128, 512);
  // MLP f2 (182 padded to 192 to stay on WMMA)
  gemm_wmma_f32<1><<<32 * 8, 32, 0, stream>>>(B1, Wf2a, bf2a, B0, 512, 128, 128);
  gemm_wmma_f32<0><<<32 * 12, 32, 0, stream>>>(B0, W2p, b2p, B2, 512, 192, 128);
  // final scatter into gt_lm copy
  scatter_out<<<524, 256, 0, stream>>>(gt, B2, out);
}